// GRUModel_32229434589511
// MI455X (gfx1250) — compile-verified
//
#include <hip/hip_runtime.h>
#include <cmath>

// ---------------------------------------------------------------------------
// GRU (H=32) over x[B=4096,T=60,F=158], fp32, for MI455X (gfx1250, wave32).
// Stage 1: gi[t][b][g] = sum_f x[b,t,f] * w_ih[g,f]   (WMMA f32 16x16x4)
// Stage 2: sequential T-scan of the GRU cell (WMMA for h @ w_hh^T) + MLP head
// Both stages are HBM-bound (250 MB / 94 MB traffic), so full-precision f32
// WMMA is the right tool: it matches reference numerics and the tensor path
// trivially covers the required ~1 TFLOP/s.
// ---------------------------------------------------------------------------

typedef __attribute__((ext_vector_type(2))) float v2f;
typedef __attribute__((ext_vector_type(8))) float v8f;

#define B_SZ 4096
#define T_SZ 60
#define F_SZ 158
#define F_PAD 160      // K padded to multiple of 4 (pad cols zeroed in LDS)
#define G3 96          // 3*H
#define H_SZ 32

__device__ __forceinline__ float sigmoid_f(float x) {
    return 1.0f / (1.0f + __expf(-x));
}

// ---------------------------------------------------------------------------
// Kernel 1: input projection GEMM.
// grid = 1920 blocks x 256 threads (8 waves). Wave computes a 16x96 tile of
// gi for 16 consecutive rows m of the flattened [B*T, F] x matrix.
// w_ih is staged in LDS ([96][160], zero-padded cols 158/159).
// ---------------------------------------------------------------------------
extern "C" __global__ __launch_bounds__(256)
void gru_gemm_in(const float* __restrict__ x, const float* __restrict__ w_ih,
                 float* __restrict__ gi)
{
    __shared__ float w_lds[G3 * F_PAD];   // 61440 B < 64 KB
    const int tid = threadIdx.x;
    for (int i = tid; i < G3 * F_PAD; i += 256) {
        const int row = i / F_PAD, col = i % F_PAD;
        w_lds[i] = (col < F_SZ) ? w_ih[row * F_SZ + col] : 0.0f;
    }
    __syncthreads();

    const int lane = tid & 31;
    const int lo   = lane & 15;      // N index within tile / M row for A
    const int hi   = lane >> 4;      // K-pair select (A/B), M-half select (C/D)
    const int wave = tid >> 5;
    const long m0  = (long)blockIdx.x * 128 + wave * 16;   // 1920*128 = 245760 exact

    v8f acc[6] = {};                 // 6 n-tiles of 16 -> 96 cols
    const float* xrow = x + (m0 + lo) * F_SZ;

    for (int kt = 0; kt < 40; ++kt) {
        const int c = kt * 4 + 2 * hi;      // K columns {c, c+1} for this lane
        v2f a = {};
        if (c < F_SZ) {                      // only false for hi lanes at kt==39
            a.x = xrow[c];
            a.y = xrow[c + 1];
        }
        #pragma unroll
        for (int nt = 0; nt < 6; ++nt) {
            const int n = nt * 16 + lo;      // B[k][n] = w_ih[n][k]
            v2f b;
            b.x = w_lds[n * F_PAD + c];
            b.y = w_lds[n * F_PAD + c + 1];
            acc[nt] = __builtin_amdgcn_wmma_f32_16x16x4_f32(
                false, a, false, b, (short)0, acc[nt], false, false);
        }
    }

    // Store C-layout tile: VGPR r -> row M = hi*8 + r, col = nt*16 + lo.
    // gi layout [T][B][96]: row m -> (t = m % 60, b = m / 60).
    #pragma unroll
    for (int r = 0; r < 8; ++r) {
        const long m = m0 + hi * 8 + r;
        const long t = m % T_SZ;
        const long b = m / T_SZ;
        float* orow = gi + (t * B_SZ + b) * G3;
        #pragma unroll
        for (int nt = 0; nt < 6; ++nt) {
            orow[nt * 16 + lo] = acc[nt][r];
        }
    }
}

// ---------------------------------------------------------------------------
// Kernel 2: sequential GRU scan + fused MLP head.
// grid = 128 blocks x 64 threads (2 waves). Wave owns 16 batch rows.
// h[16][32] lives in wave-private LDS (row stride 34 -> conflict-free A loads);
// w_hh B-fragments are register-resident (loop-invariant over T).
// Per step: 48 x v_wmma_f32_16x16x4_f32 computes gh = h @ w_hh^T.
// ---------------------------------------------------------------------------
extern "C" __global__ __launch_bounds__(64)
void gru_recurrence(const float* __restrict__ gi, const float* __restrict__ w_hh,
                    const float* __restrict__ w1, const float* __restrict__ b1,
                    const float* __restrict__ w2, float* __restrict__ out)
{
    __shared__ float h_lds[2 * 16 * 34];
    const int tid = threadIdx.x;
    for (int i = tid; i < 2 * 16 * 34; i += 64) h_lds[i] = 0.0f;  // h0 = 0
    __syncthreads();

    const int lane = tid & 31;
    const int lo   = lane & 15;
    const int hi   = lane >> 4;
    const int wave = tid >> 5;
    const int bb   = blockIdx.x * 32 + wave * 16;   // batch base for this wave
    float* hw = h_lds + wave * (16 * 34);

    // Preload w_hh B-fragments: B[k][n] = w_hh[n*32 + k], contiguous pairs.
    v2f bf[6][8];
    #pragma unroll
    for (int nt = 0; nt < 6; ++nt) {
        const int n = nt * 16 + lo;
        #pragma unroll
        for (int kt = 0; kt < 8; ++kt) {
            const int k = kt * 4 + 2 * hi;
            bf[nt][kt].x = w_hh[n * H_SZ + k];
            bf[nt][kt].y = w_hh[n * H_SZ + k + 1];
        }
    }

    v8f hc[2] = {};   // h in C-layout: 2 n-tiles covering cols 0..31

    for (int t = 0; t < T_SZ; ++t) {
        // gh = h @ w_hh^T via WMMA; A fragments streamed from LDS h.
        v8f acc[6] = {};
        #pragma unroll
        for (int kt = 0; kt < 8; ++kt) {
            const int c = kt * 4 + 2 * hi;
            v2f a;
            a.x = hw[lo * 34 + c];
            a.y = hw[lo * 34 + c + 1];
            #pragma unroll
            for (int nt = 0; nt < 6; ++nt) {
                acc[nt] = __builtin_amdgcn_wmma_f32_16x16x4_f32(
                    false, a, false, bf[nt][kt], (short)0, acc[nt], false, false);
            }
        }

        // Gates. gi_t tile rows: two 64B segments per lane-load (coalesced-ish).
        const float* girow = gi + ((long)t * B_SZ + bb) * G3;
        #pragma unroll
        for (int nt = 0; nt < 2; ++nt) {     // h columns 0..15 / 16..31
            v8f ir, iz, inn;
            #pragma unroll
            for (int r = 0; r < 8; ++r) {
                const int row = hi * 8 + r;
                const float* p = girow + (long)row * G3 + lo;
                ir[r]  = p[(nt + 0) * 16];   // i_r: cols  0..31
                iz[r]  = p[(nt + 2) * 16];   // i_z: cols 32..63
                inn[r] = p[(nt + 4) * 16];   // i_n: cols 64..95
            }
            #pragma unroll
            for (int r = 0; r < 8; ++r) {
                const float rg = sigmoid_f(ir[r] + acc[nt][r]);
                const float zg = sigmoid_f(iz[r] + acc[nt + 2][r]);
                const float ng = tanhf(inn[r] + rg * acc[nt + 4][r]);
                hc[nt][r] = (1.0f - zg) * ng + zg * hc[nt][r];
            }
            // C-layout -> LDS so next step's A fragments see updated h.
            #pragma unroll
            for (int r = 0; r < 8; ++r) {
                hw[(hi * 8 + r) * 34 + nt * 16 + lo] = hc[nt][r];
            }
        }
        __syncthreads();   // order LDS store -> next-iter cross-lane A load
    }

    // Fused MLP head: out[b] = relu(h @ w1^T + b1) @ w2^T. Lanes 0..15 each
    // handle one of the wave's 16 rows; weights broadcast from cache.
    if (lane < 16) {
        float hrow[H_SZ];
        #pragma unroll
        for (int k = 0; k < H_SZ; ++k) hrow[k] = hw[lane * 34 + k];
        float y = 0.0f;
        for (int j = 0; j < 16; ++j) {
            float a = b1[j];
            #pragma unroll
            for (int k = 0; k < H_SZ; ++k) a += hrow[k] * w1[j * H_SZ + k];
            y += w2[j] * fmaxf(a, 0.0f);
        }
        out[bb + lane] = y;
    }
}

// ---------------------------------------------------------------------------
extern "C" void kernel_launch(void* const* d_in, const int* in_sizes, int n_in,
                              void* d_out, int out_size, void* d_ws, size_t ws_size,
                              hipStream_t stream) {
    const float* x    = (const float*)d_in[0];   // [4096,60,158]
    const float* w_ih = (const float*)d_in[1];   // [96,158]
    const float* w_hh = (const float*)d_in[2];   // [96,32]
    const float* w1   = (const float*)d_in[3];   // [16,32]
    const float* b1   = (const float*)d_in[4];   // [16]
    const float* w2   = (const float*)d_in[5];   // [1,16]
    float* out = (float*)d_out;                  // [4096]

    float* gi = (float*)d_ws;                    // [60][4096][96] f32 = 94.4 MB

    gru_gemm_in<<<1920, 256, 0, stream>>>(x, w_ih, gi);
    gru_recurrence<<<128, 64, 0, stream>>>(gi, w_hh, w1, b1, w2, out);
}